// MLPPredictor_15676630630926
// MI455X (gfx1250) — compile-verified
//
#include <hip/hip_runtime.h>

// Hypernetwork MLP: one private 4-layer MLP per (s,b), weights packed in z.
// HBM-bound (0.5 FLOP/byte, ~1.49 GB of z streamed once at 23.3 TB/s -> ~64us
// floor). fp32 WMMA (16x16x4) is numerically exact and compute is ~10x under
// the matrix ceiling, so the kernel is organized purely around a strictly
// linear, non-temporal streaming read of each sample's contiguous z slice:
// k-group outer / j-tile inner with per-tile accumulators, so rows k0..k0+3
// (2KB contiguous) are consumed whole before advancing.

typedef __attribute__((ext_vector_type(2))) float v2f;
typedef __attribute__((ext_vector_type(8))) float v8f;

#define X_DIM   64
#define Y_DIM   32
#define H_DIM   128
#define Z_DIM   45472
#define NSAMP   8192          // S*B = 128*64
#define WAVES_PER_BLOCK 8
#define BLOCK   (WAVES_PER_BLOCK * 32)

// One layer of the per-sample MLP, executed by one wave32.
//  w    : weight base for this sample/layer, row-major [NIN][NOUT]
//  bias : NOUT floats
//  cur  : LDS, current activations (NIN floats)
//  nxt  : LDS, next activations (NOUT floats) -- unused if LAST
//  gout : global output (LAST only)
template <int NIN, int NOUT, bool RELU, bool LAST>
__device__ __forceinline__ void hyper_layer(const float* __restrict__ w,
                                            const float* __restrict__ bias,
                                            const float* cur, float* nxt,
                                            float* __restrict__ gout,
                                            int l15, int half, int lane)
{
    constexpr int NJT = NOUT / 16;      // output tiles of 16
    v8f acc[NJT];
#pragma unroll
    for (int jt = 0; jt < NJT; ++jt)
        acc[jt] = (v8f){0.f, 0.f, 0.f, 0.f, 0.f, 0.f, 0.f, 0.f};

    // This lane's column base within the B-tile:
    //   W[k0 + 2*half + r][jt*16 + l15], r = 0,1
    const float* wp = w + l15 + (size_t)(2 * half) * NOUT;
    const float* cp = cur + 2 * half;

    for (int k0 = 0; k0 < NIN; k0 += 4) {
        // A (16x4 f32 layout): lanes 0-15 hold K0,K1; lanes 16-31 hold K2,K3.
        // Activation broadcast from per-wave LDS (uniform per half); built
        // once per k-group, shared by all NJT output tiles.
        v2f a;
        a.x = cp[k0];
        a.y = cp[k0 + 1];
        const float* wk = wp + (size_t)k0 * NOUT;
#pragma unroll
        for (int jt = 0; jt < NJT; ++jt) {
            // B (4x16 f32): VGPR0 = K0/K2 rows, VGPR1 = K1/K3 rows, N = l15.
            // Together the NJT tiles consume rows k0..k0+3 completely
            // (perfectly sequential 2KB), so NT is exactly right: every
            // byte is read once and never wanted again.
            v2f b;
            b.x = __builtin_nontemporal_load(wk + jt * 16);
            b.y = __builtin_nontemporal_load(wk + jt * 16 + NOUT);
            // D[m,n] = sum_k act[k0+k] * W[k0+k][j0+n]  (same for every m)
            acc[jt] = __builtin_amdgcn_wmma_f32_16x16x4_f32(
                false, a, false, b, (short)0, acc[jt], false, false);
        }
    }

#pragma unroll
    for (int jt = 0; jt < NJT; ++jt) {
        // acc[jt][0], lanes 0-15 == out[jt*16 .. jt*16+15]
        float v = acc[jt][0] + bias[jt * 16 + l15];
        if (RELU) v = v > 0.f ? v : 0.f;
        if (lane < 16) {
            if (LAST) {
                __builtin_nontemporal_store(v, gout + jt * 16 + l15);
            } else {
                nxt[jt * 16 + l15] = v;
            }
        }
    }
}

__global__ __launch_bounds__(BLOCK) void mlp_hyper_kernel(
    const float* __restrict__ x,   // [NSAMP][X_DIM]
    const float* __restrict__ z,   // [NSAMP][Z_DIM]
    float* __restrict__ out)       // [NSAMP][Y_DIM]
{
    __shared__ float smem[WAVES_PER_BLOCK][2][H_DIM];

    const int lane = threadIdx.x & 31;
    const int wave = threadIdx.x >> 5;
    const int l15  = lane & 15;
    const int half = (lane >> 4) & 1;

    const int sample = blockIdx.x * WAVES_PER_BLOCK + wave;
    if (sample >= NSAMP) return;  // wave-uniform

    const float* xp = x + (size_t)sample * X_DIM;
    const float* zp = z + (size_t)sample * Z_DIM;
    float* gout     = out + (size_t)sample * Y_DIM;

    float* buf0 = smem[wave][0];
    float* buf1 = smem[wave][1];

    // Stage the 64-float input vector into per-wave LDS (coalesced 128B loads).
    buf0[lane]      = xp[lane];
    buf0[lane + 32] = xp[lane + 32];
    // Same-wave DS ops are in-order; compiler inserts s_wait_dscnt before use.

    // z layout per sample:
    //   w0   [0,8192)           b0 [8192,8320)
    //   w1   [8320,24704)       b1 [24704,24832)
    //   w2   [24832,41216)      b2 [41216,41344)
    //   wlast[41344,45440)      blast [45440,45472)
    const float* p = zp;
    hyper_layer<X_DIM, H_DIM, true, false>(p, p + X_DIM * H_DIM,
                                           buf0, buf1, nullptr, l15, half, lane);
    p += X_DIM * H_DIM + H_DIM;
    hyper_layer<H_DIM, H_DIM, true, false>(p, p + H_DIM * H_DIM,
                                           buf1, buf0, nullptr, l15, half, lane);
    p += H_DIM * H_DIM + H_DIM;
    hyper_layer<H_DIM, H_DIM, true, false>(p, p + H_DIM * H_DIM,
                                           buf0, buf1, nullptr, l15, half, lane);
    p += H_DIM * H_DIM + H_DIM;
    hyper_layer<H_DIM, Y_DIM, false, true>(p, p + H_DIM * Y_DIM,
                                           buf1, nullptr, gout, l15, half, lane);
}

extern "C" void kernel_launch(void* const* d_in, const int* in_sizes, int n_in,
                              void* d_out, int out_size, void* d_ws, size_t ws_size,
                              hipStream_t stream) {
    const float* x = (const float*)d_in[0];   // (S,B,X_DIM) f32
    const float* z = (const float*)d_in[1];   // (S,B,Z_DIM) f32
    float* out = (float*)d_out;               // (S,B,Y_DIM) f32
    (void)in_sizes; (void)n_in; (void)out_size; (void)d_ws; (void)ws_size;

    dim3 grid(NSAMP / WAVES_PER_BLOCK);       // 1024 blocks x 256 threads = 8192 waves
    mlp_hyper_kernel<<<grid, dim3(BLOCK), 0, stream>>>(x, z, out);
}